// _PyGLaplacianRefiner_17910013624467
// MI455X (gfx1250) — compile-verified
//
#include <hip/hip_runtime.h>
#include <hip/hip_bf16.h>
#include <math.h>

typedef float v2f __attribute__((ext_vector_type(2)));
typedef float v8f __attribute__((ext_vector_type(8)));

#define BATCH 16

// Branchless erf (Abramowitz-Stegun 7.1.26, max abs err ~1.5e-7):
// erf(x) = sign(x) * (1 - poly(t) * exp(-x^2)),  t = 1/(1 + 0.3275911*|x|)
// Straight-line: no EXEC divergence between WMMAs; uses v_rcp_f32 + v_exp_f32.
__device__ __forceinline__ float fast_erf(float x) {
    float ax = fabsf(x);
    float t  = __builtin_amdgcn_rcpf(fmaf(0.3275911f, ax, 1.0f));
    float p  = fmaf(1.061405429f, t, -1.453152027f);
    p = fmaf(p, t, 1.421413741f);
    p = fmaf(p, t, -0.284496736f);
    p = fmaf(p, t, 0.254829592f);
    p = p * t;
    float e = __expf(-ax * ax);          // underflows to 0 for large |x| -> erf saturates
    float y = fmaf(-p, e, 1.0f);
    return copysignf(y, x);
}

__device__ __forceinline__ float gelu_erf(float t) {
    // exact-form GELU: 0.5*t*(1+erf(t/sqrt(2)))
    return 0.5f * t * (1.0f + fast_erf(t * 0.70710678118654752440f));
}

// ---------------- zero the accumulators (agg[N*16] ++ deg[N]) ----------------
__global__ void k_zero(float* __restrict__ p, int count) {
    int t = blockIdx.x * blockDim.x + threadIdx.x;
    if (t < count) p[t] = 0.0f;
}

// ---------------- transpose mu [B,N] -> muT [N,B] ----------------
__global__ void k_transpose(const float* __restrict__ mu, float* __restrict__ muT,
                            int N, int total) {
    int t = blockIdx.x * blockDim.x + threadIdx.x;  // t = b*N + n
    if (t >= total) return;
    int b = t / N;
    int n = t - b * N;
    muT[n * BATCH + b] = mu[t];
}

// ---------------- edge scatter: agg[row,:] += muT[col,:]; deg[row] += 1 ----------------
// 16 consecutive lanes handle the 16 batch elements of one edge -> 64B-coalesced
// gathers from muT and 64B-coalesced atomic bursts into L2-resident agg.
__global__ void k_scatter(const int* __restrict__ e_row, const int* __restrict__ e_col,
                          const float* __restrict__ muT,
                          float* __restrict__ agg, float* __restrict__ deg,
                          long long total) {
    long long t = (long long)blockIdx.x * blockDim.x + threadIdx.x;
    if (t >= total) return;
    int e = (int)(t >> 4);
    int b = (int)(t & 15);
    int r = e_row[e];
    int c = e_col[e];
    float v = muT[c * BATCH + b];
    unsafeAtomicAdd(&agg[r * BATCH + b], v);
    if (b == 0) unsafeAtomicAdd(&deg[r], 1.0f);
}

// ---------------- MLP 1->64->1 with exact-form GELU; h@W2 via V_WMMA_F32_16X16X4_F32 ----------------
// One wave32 produces 16 output scalars. A-matrix (16x4, f32): lanes 0-15 carry
// K={4k,4k+1} in VGPR0/1, lanes 16-31 carry K={4k+2,4k+3} (ISA layout). B (4x16)
// broadcasts W2[k] across all 16 columns, so every column of D equals y_tile.
// 16 accumulating WMMAs cover H=64. Column 0 extracted by lanes 0 (rows 0-7)
// and 16 (rows 8-15) per the documented C/D VGPR layout. Control flow is fully
// uniform (branchless GELU), so EXEC stays all-ones through every WMMA.
__global__ void k_mlp_wmma(const float* __restrict__ agg, const float* __restrict__ deg,
                           const float* __restrict__ W1, const float* __restrict__ b1,
                           const float* __restrict__ W2, const float* __restrict__ b2,
                           float* __restrict__ y, int N) {
    __shared__ float sW1[64], sB1[64], sW2[64];
    int tid = threadIdx.x;
    if (tid < 64) {
        sW1[tid] = W1[tid];
        sB1[tid] = b1[tid];
        sW2[tid] = W2[tid];
    }
    __syncthreads();
    const float b2v = b2[0];

    int wave  = (blockIdx.x * blockDim.x + tid) >> 5;  // global wave id
    int lane  = tid & 31;
    int sbase = wave * 16;                              // 16 consecutive flat outputs
    int M     = lane & 15;                              // A-matrix row handled by this lane
    int s     = sbase + M;                              // flat index = b*N + n
    int b     = s / N;
    int n     = s - b * N;

    float d = deg[n];
    d = (d < 1.0f) ? 1.0f : d;
    float x = agg[n * BATCH + b] / d;                   // out[b,n]

    int koff = (lane < 16) ? 0 : 2;                     // K-pair within each 4-chunk

    v8f c = {};
#pragma unroll
    for (int k = 0; k < 16; ++k) {
        int kk = 4 * k + koff;
        float a0 = gelu_erf(fmaf(x, sW1[kk],     sB1[kk]));
        float a1 = gelu_erf(fmaf(x, sW1[kk + 1], sB1[kk + 1]));
        v2f A  = {a0, a1};
        v2f Bv = {sW2[kk], sW2[kk + 1]};                // B row K broadcast over N
        c = __builtin_amdgcn_wmma_f32_16x16x4_f32(
                /*neg_a=*/false, A, /*neg_b=*/false, Bv,
                /*c_mod=*/(short)0, c, /*reuse_a=*/false, /*reuse_b=*/false);
    }

    // Column N=0 of D: lane 0 holds rows 0-7 in c[0..7], lane 16 holds rows 8-15.
    if ((lane & 15) == 0) {
        float* yp = y + sbase + ((lane < 16) ? 0 : 8);
        yp[0] = c[0] + b2v;
        yp[1] = c[1] + b2v;
        yp[2] = c[2] + b2v;
        yp[3] = c[3] + b2v;
        yp[4] = c[4] + b2v;
        yp[5] = c[5] + b2v;
        yp[6] = c[6] + b2v;
        yp[7] = c[7] + b2v;
    }
}

extern "C" void kernel_launch(void* const* d_in, const int* in_sizes, int n_in,
                              void* d_out, int out_size, void* d_ws, size_t ws_size,
                              hipStream_t stream) {
    const float* mu   = (const float*)d_in[0];   // [B, N]
    const int*   eidx = (const int*)d_in[1];     // [2, E]
    const float* W1   = (const float*)d_in[2];   // [1, 64]
    const float* b1   = (const float*)d_in[3];   // [64]
    const float* W2   = (const float*)d_in[4];   // [64, 1]
    const float* b2   = (const float*)d_in[5];   // [1]
    float*       y    = (float*)d_out;           // [B, N]

    const int N = in_sizes[0] / BATCH;           // 100000
    const int E = in_sizes[1] / 2;               // 3200000
    const int* e_row = eidx;
    const int* e_col = eidx + E;

    // workspace layout: agg[N*16] | deg[N] | muT[N*16]
    float* agg = (float*)d_ws;
    float* deg = agg + (size_t)N * BATCH;
    float* muT = deg + N;

    const int TPB = 256;

    // 1) zero agg + deg (contiguous N*17 floats)
    {
        int count = N * (BATCH + 1);
        k_zero<<<(count + TPB - 1) / TPB, TPB, 0, stream>>>(agg, count);
    }
    // 2) transpose mu -> muT
    {
        int total = BATCH * N;
        k_transpose<<<(total + TPB - 1) / TPB, TPB, 0, stream>>>(mu, muT, N, total);
    }
    // 3) edge scatter (16 lanes per edge)
    {
        long long total = (long long)E * BATCH;
        long long blocks = (total + TPB - 1) / TPB;
        k_scatter<<<(unsigned)blocks, TPB, 0, stream>>>(e_row, e_col, muT, agg, deg, total);
    }
    // 4) MLP with WMMA: one wave per 16 outputs
    {
        int waves  = (BATCH * N) / 16;           // 100000 (exact)
        int blocks = waves / (TPB / 32);         // 12500 (exact)
        k_mlp_wmma<<<blocks, TPB, 0, stream>>>(agg, deg, W1, b1, W2, b2, y, N);
    }
}